// QuantizedBayesNetClassifier_37194416783931
// MI455X (gfx1250) — compile-verified
//
#include <hip/hip_runtime.h>
#include <math.h>

// Problem dims (fixed by the reference).
#define C_DIM 128
#define U_DIM 256
#define F_DIM 64
#define B_TILE 256
#define TBL_FLOATS (U_DIM * C_DIM)        // 32768 floats per feature table
#define TBL_BYTES  (TBL_FLOATS * 4)       // 131072 bytes = 128 KB
#define SMEM_BYTES (2 * TBL_BYTES + B_TILE * 4)  // double buffer + index tile = 263168 B (< 320 KB LDS)

#define QMAX 15.9375f  // 2^4 - 2^-4

#if defined(__HIP_DEVICE_COMPILE__)
#if __has_builtin(__builtin_amdgcn_tensor_load_to_lds) && __has_builtin(__builtin_amdgcn_s_wait_tensorcnt)
#define USE_TDM 1
#endif
#endif

// ---------------- Preprocess: quantized class log-probs ----------------
__global__ void quantize_class(const float* __restrict__ cl, float* __restrict__ clq) {
  __shared__ float red[C_DIM];
  int t = threadIdx.x;
  float v = cl[t];
  red[t] = v;
  __syncthreads();
  for (int s = C_DIM / 2; s > 0; s >>= 1) {
    if (t < s) red[t] = fmaxf(red[t], red[t + s]);
    __syncthreads();
  }
  float m = red[0];
  __syncthreads();
  red[t] = expf(v - m);
  __syncthreads();
  for (int s = C_DIM / 2; s > 0; s >>= 1) {
    if (t < s) red[t] += red[t + s];
    __syncthreads();
  }
  float lse = m + logf(red[0]);
  float y = -(v - lse);                                  // >= 0
  float q = fminf(fmaxf(rintf(y * 16.0f) * 0.0625f, 0.0f), QMAX);
  clq[t] = -q;
}

// ---------------- Preprocess: quantized feature log-probs ----------------
// One block per feature; one thread per class column; logsumexp over U (online).
__global__ void quantize_features(const float* __restrict__ fl, float* __restrict__ flq) {
  int f = blockIdx.x;
  int c = threadIdx.x;  // 0..127
  const float* base = fl + (size_t)f * TBL_FLOATS + c;
  float m = -INFINITY, s = 0.0f;
  for (int u = 0; u < U_DIM; ++u) {
    float v = base[(size_t)u * C_DIM];
    if (v > m) { s = s * expf(m - v) + 1.0f; m = v; }
    else       { s += expf(v - m); }
  }
  float lse = m + logf(s);
  float* ob = flq + (size_t)f * TBL_FLOATS + c;
  for (int u = 0; u < U_DIM; ++u) {
    float v = base[(size_t)u * C_DIM] - lse;
    float y = -v;
    float q = fminf(fmaxf(rintf(y * 16.0f) * 0.0625f, 0.0f), QMAX);
    ob[(size_t)u * C_DIM] = -q;
  }
}

// ---------------- Main: gather-accumulate with TDM-staged LDS tables ----------------
#if USE_TDM
typedef unsigned int u32x4 __attribute__((ext_vector_type(4)));
typedef int i32x8 __attribute__((ext_vector_type(8)));
typedef int i32x4 __attribute__((ext_vector_type(4)));

// Build a D# for a linear 128 KB (32768 x f32) tile copy: global -> LDS.
__device__ __forceinline__ void tdm_issue(unsigned lds_off, const void* gptr) {
  unsigned long long ga = (unsigned long long)(size_t)gptr;
  u32x4 g0;
  g0[0] = 1u;                                   // count=1 (valid), user mode, no gather
  g0[1] = lds_off;                              // LDS byte address of tile start
  g0[2] = (unsigned)ga;                         // global_addr[31:0]
  g0[3] = (unsigned)(ga >> 32) | (2u << 30);    // global_addr[56:32] | type=2 (image)
  i32x8 g1;
  g1[0] = 0x00020000;                           // workgroup_mask=0, data_size=2 (4B)
  g1[1] = (int)0x80000000u;                     // tensor_dim0 = 32768 (low 16 @ bits 63:48)
  g1[2] = 0x00010000;                           // tensor_dim0 hi16=0; tensor_dim1 = 1
  g1[3] = (int)0x80000000u;                     // tile_dim0 = 32768 (@ bits 127:112)
  g1[4] = 1;                                    // tile_dim1 = 1, tile_dim2 = 0
  g1[5] = 32768;                                // tensor_dim0_stride low32
  g1[6] = (int)0x80000000u;                     // stride0 hi16=0; tensor_dim1_stride low16 = 32768
  g1[7] = 0;                                    // tensor_dim1_stride hi bits = 0
  i32x4 z4 = {0, 0, 0, 0};
  i32x8 z8 = {0, 0, 0, 0, 0, 0, 0, 0};
  // clang-23 / therock signature: (g0, g1, g2, g3, g4, cpol)
  __builtin_amdgcn_tensor_load_to_lds(g0, g1, z4, z4, z8, 0);
}
#endif

__global__ __launch_bounds__(1024)
void bayes_accumulate(const int* __restrict__ x,
                      const float* __restrict__ clq,
                      const float* __restrict__ flq,
                      float* __restrict__ out) {
  extern __shared__ char smem[];
  float* buf0 = (float*)smem;
  float* buf1 = (float*)(smem + TBL_BYTES);
  int* sx = (int*)(smem + 2 * TBL_BYTES);

  const int tid = threadIdx.x;
  const int lane = tid & 31;   // class group: c = lane*4 .. lane*4+3
  const int wv = tid >> 5;     // 0..31: sample-group within tile
  const int b0 = blockIdx.x * B_TILE;

  // init accumulators with quantized class prior
  float4 cinit = ((const float4*)clq)[lane];
  float4 acc[8];
#pragma unroll
  for (int j = 0; j < 8; ++j) acc[j] = cinit;

  // prologue: stage feature 0 table into buf0
#if USE_TDM
  if (tid < 32) tdm_issue((unsigned)(size_t)buf0, flq);
#else
  {
    const float4* src = (const float4*)flq;
    float4* dst = (float4*)buf0;
#pragma unroll
    for (int k = 0; k < 8; ++k) dst[tid + 1024 * k] = src[tid + 1024 * k];
  }
#endif

  for (int f = 0; f < F_DIM; ++f) {
    __syncthreads();  // (1) previous iteration's LDS reads are done; safe to overwrite

    // stage this feature's sample indices
    if (tid < B_TILE) sx[tid] = x[(size_t)(b0 + tid) * F_DIM + f];

    float* nxt = ((f + 1) & 1) ? buf1 : buf0;
#if USE_TDM
    if (tid < 32) {
      if (f + 1 < F_DIM) {
        tdm_issue((unsigned)(size_t)nxt, flq + (size_t)(f + 1) * TBL_FLOATS);
        __builtin_amdgcn_s_wait_tensorcnt(1);   // table f has landed (in-order TDM)
      } else {
        __builtin_amdgcn_s_wait_tensorcnt(0);
      }
    }
#else
    if (f + 1 < F_DIM) {
      const float4* src = (const float4*)(flq + (size_t)(f + 1) * TBL_FLOATS);
      float4* dst = (float4*)nxt;
#pragma unroll
      for (int k = 0; k < 8; ++k) dst[tid + 1024 * k] = src[tid + 1024 * k];
    }
#endif
    __syncthreads();  // (2) table f + indices visible to all waves

    const float* tb = (f & 1) ? buf1 : buf0;
#pragma unroll
    for (int j = 0; j < 8; ++j) {
      int idx = sx[wv + 32 * j];                        // wave-uniform row index
      const float4* row = (const float4*)(tb + (size_t)idx * C_DIM);
      float4 v = row[lane];                             // 32 lanes x 16B: conflict-free ds_load_b128
      acc[j].x += v.x; acc[j].y += v.y; acc[j].z += v.z; acc[j].w += v.w;
    }
  }

  // write out: contiguous 512 B per wave-row
#pragma unroll
  for (int j = 0; j < 8; ++j) {
    int b = b0 + wv + 32 * j;
    ((float4*)out)[(size_t)b * (C_DIM / 4) + lane] = acc[j];
  }
}

extern "C" void kernel_launch(void* const* d_in, const int* in_sizes, int n_in,
                              void* d_out, int out_size, void* d_ws, size_t ws_size,
                              hipStream_t stream) {
  const float* class_logits = (const float*)d_in[0];
  const float* feature_logits = (const float*)d_in[1];
  const int* x = (const int*)d_in[2];
  float* out = (float*)d_out;

  float* clq = (float*)d_ws;          // [128]
  float* flq = clq + C_DIM;           // [F,U,C] quantized tables (16B-aligned: 512 B offset)

  quantize_class<<<1, C_DIM, 0, stream>>>(class_logits, clq);
  quantize_features<<<F_DIM, C_DIM, 0, stream>>>(feature_logits, flq);

  int B = in_sizes[2] / F_DIM;        // 131072
  (void)hipFuncSetAttribute(reinterpret_cast<const void*>(bayes_accumulate),
                            hipFuncAttributeMaxDynamicSharedMemorySize, (int)SMEM_BYTES);
  bayes_accumulate<<<B / B_TILE, 1024, SMEM_BYTES, stream>>>(x, clq, flq, out);
}